// LSTM_Model_27032524161517
// MI455X (gfx1250) — compile-verified
//
#include <hip/hip_runtime.h>
#include <math.h>

#define BATCH 1024
#define TT    64
#define INW   64
#define HH    51
#define NMEM  128
#define MMEM  20
#define EPSF  1e-6f

#define BT    16      // batch tile per block
#define TPB   256     // 8 wave32
#define NWAVE 8

typedef float v2f __attribute__((ext_vector_type(2)));
typedef float v8f __attribute__((ext_vector_type(8)));

// ---- packed-weight workspace layout (floats) -------------------------------
// B-fragments stored as [tile][kstep][lane(0..31)][2] so a wave reads one
// contiguous 256B block per k-step: lane reads (n = tile*16 + (lane&15),
// kk = kstep*4 + (lane>>4)*2 + {0,1}).
constexpr int G1_TILES = 13, G1_KS = 29;          // layer-1 gates, K=116
constexpr int G2_TILES = 13, G2_KS = 26;          // layer-2 gates, K=104
constexpr int GH_TILES = 6,  GH_KS = 13;          // heads,         K=52
constexpr int G1_FLOATS = G1_TILES * G1_KS * 64;  // 24128
constexpr int G2_FLOATS = G2_TILES * G2_KS * 64;  // 21632
constexpr int GH_FLOATS = GH_TILES * GH_KS * 64;  //  4992
constexpr int OFF_G2 = G1_FLOATS;
constexpr int OFF_GH = OFF_G2 + G2_FLOATS;
constexpr int OFF_B1 = OFF_GH + GH_FLOATS;
constexpr int OFF_B2 = OFF_B1 + 208;
constexpr int OFF_BH = OFF_B2 + 208;
constexpr int WS_FLOATS = OFF_BH + 96;            // 51264 floats

__device__ __forceinline__ float sigm(float x) { return 1.f / (1.f + expf(-x)); }
__device__ __forceinline__ float softplusf(float x) { return (x > 20.f) ? x : log1pf(expf(x)); }

// ---- padded/concatenated weight element rules ------------------------------
__device__ __forceinline__ float wval1(const float* Wih1, const float* Whh1, int n, int k) {
  if (n >= 204) return 0.f;
  if (k < 64) return Wih1[n * 64 + k];
  const int kh = k - 64;
  return (kh < 51) ? Whh1[n * 51 + kh] : 0.f;
}
__device__ __forceinline__ float wval2(const float* Wih2, const float* Whh2, int n, int k) {
  if (n >= 204) return 0.f;
  if (k < 52) return (k < 51) ? Wih2[n * 51 + k] : 0.f;
  const int kh = k - 52;
  return (kh < 51) ? Whh2[n * 51 + kh] : 0.f;
}
__device__ __forceinline__ float wvalh(const float* Wread, const float* Wwrite, int n, int k) {
  if (k >= 51) return 0.f;
  if (n < 26) return Wread[n * 51 + k];
  if (n < 92) return Wwrite[(n - 26) * 51 + k];
  return 0.f;
}

// ---- one-shot pack kernel: weights -> WMMA B-fragment order in d_ws --------
__global__ void pack_weights(const float* __restrict__ Wih1, const float* __restrict__ Whh1,
                             const float* __restrict__ Wih2, const float* __restrict__ Whh2,
                             const float* __restrict__ Wread, const float* __restrict__ Wwrite,
                             const float* __restrict__ bih1, const float* __restrict__ bhh1,
                             const float* __restrict__ bih2, const float* __restrict__ bhh2,
                             const float* __restrict__ bread, const float* __restrict__ bwrite,
                             float* __restrict__ ws) {
  const int idx = blockIdx.x * blockDim.x + threadIdx.x;
  if (idx >= WS_FLOATS) return;
  float v;
  if (idx < OFF_G2) {
    const int tile = idx / (G1_KS * 64), rem = idx % (G1_KS * 64);
    const int kstep = rem / 64, pos = rem % 64, lane = pos >> 1, e = pos & 1;
    v = wval1(Wih1, Whh1, tile * 16 + (lane & 15), kstep * 4 + ((lane >> 4) << 1) + e);
  } else if (idx < OFF_GH) {
    const int j = idx - OFF_G2;
    const int tile = j / (G2_KS * 64), rem = j % (G2_KS * 64);
    const int kstep = rem / 64, pos = rem % 64, lane = pos >> 1, e = pos & 1;
    v = wval2(Wih2, Whh2, tile * 16 + (lane & 15), kstep * 4 + ((lane >> 4) << 1) + e);
  } else if (idx < OFF_B1) {
    const int j = idx - OFF_GH;
    const int tile = j / (GH_KS * 64), rem = j % (GH_KS * 64);
    const int kstep = rem / 64, pos = rem % 64, lane = pos >> 1, e = pos & 1;
    v = wvalh(Wread, Wwrite, tile * 16 + (lane & 15), kstep * 4 + ((lane >> 4) << 1) + e);
  } else if (idx < OFF_B2) {
    const int n = idx - OFF_B1;
    v = (n < 204) ? (bih1[n] + bhh1[n]) : 0.f;
  } else if (idx < OFF_BH) {
    const int n = idx - OFF_B2;
    v = (n < 204) ? (bih2[n] + bhh2[n]) : 0.f;
  } else {
    const int n = idx - OFF_BH;
    v = (n < 26) ? bread[n] : ((n < 92) ? bwrite[n - 26] : 0.f);
  }
  ws[idx] = v;
}

// ---------------------------------------------------------------------------
// PACKED GEMM: A-fragments hoisted to registers, B-fragments are branch-free
// coalesced global_load_b64 from the pre-swizzled weights.
// ---------------------------------------------------------------------------
template <int KS>
__device__ __forceinline__ void gemm_packed(const float* act, const float* __restrict__ Wp,
                                            const float* __restrict__ biasp, int ntiles,
                                            float* outp, int outld, int wave, int lane) {
  const int l15 = lane & 15;
  const int ks  = (lane >> 4) * 2;
  v2f areg[KS];
#pragma unroll
  for (int i = 0; i < KS; ++i) {
    areg[i][0] = act[l15 * 120 + i * 4 + ks];
    areg[i][1] = act[l15 * 120 + i * 4 + ks + 1];
  }
  for (int tile = wave; tile < ntiles; tile += NWAVE) {
    const float* w = Wp + (size_t)tile * KS * 64 + lane * 2;
    v8f c = {};
#pragma unroll
    for (int i = 0; i < KS; ++i) {
      v2f b; b[0] = w[i * 64]; b[1] = w[i * 64 + 1];
      c = __builtin_amdgcn_wmma_f32_16x16x4_f32(false, areg[i], false, b, (short)0, c, false, false);
    }
    const float bn = biasp[tile * 16 + l15];
    const int mb = (lane < 16) ? 0 : 8;
#pragma unroll
    for (int r = 0; r < 8; ++r)
      outp[(mb + r) * outld + tile * 16 + l15] = c[r] + bn;
  }
}

// ---- fallback guarded loads (used only if d_ws is too small) ---------------
template <int WHICH>
__device__ __forceinline__ v2f wload(const float* __restrict__ Wa,
                                     const float* __restrict__ Wb, int n, int kk) {
  v2f r; r[0] = 0.f; r[1] = 0.f;
  if (WHICH == 1) {
    if (n < 204) {
      if (kk < 64) { const float* p = &Wa[n * 64 + kk]; r[0] = p[0]; r[1] = p[1]; }
      else { int kh = kk - 64;
        r[0] = (kh < 51) ? Wb[n * 51 + kh] : 0.f;
        r[1] = (kh + 1 < 51) ? Wb[n * 51 + kh + 1] : 0.f; }
    }
  } else if (WHICH == 2) {
    if (n < 204) {
      if (kk < 52) { r[0] = (kk < 51) ? Wa[n * 51 + kk] : 0.f;
                     r[1] = (kk + 1 < 51) ? Wa[n * 51 + kk + 1] : 0.f; }
      else { int kh = kk - 52;
        r[0] = (kh < 51) ? Wb[n * 51 + kh] : 0.f;
        r[1] = (kh + 1 < 51) ? Wb[n * 51 + kh + 1] : 0.f; }
    }
  } else {
    const float* W; int nn;
    if (n < 26) { W = Wa; nn = n; }
    else if (n < 92) { W = Wb; nn = n - 26; }
    else return r;
    r[0] = (kk < 51) ? W[nn * 51 + kk] : 0.f;
    r[1] = (kk + 1 < 51) ? W[nn * 51 + kk + 1] : 0.f;
  }
  return r;
}

template <int WHICH, int KTOT>
__device__ __forceinline__ void gemm_guarded(const float* act, int ntiles,
                                             const float* __restrict__ Wa, const float* __restrict__ Wb,
                                             const float* __restrict__ ba, const float* __restrict__ bb,
                                             float* outp, int outld, int wave, int lane) {
  constexpr int KS = KTOT / 4;
  const int l15 = lane & 15;
  const int ks  = (lane >> 4) * 2;
  v2f areg[KS];
#pragma unroll
  for (int i = 0; i < KS; ++i) {
    areg[i][0] = act[l15 * 120 + i * 4 + ks];
    areg[i][1] = act[l15 * 120 + i * 4 + ks + 1];
  }
  for (int tile = wave; tile < ntiles; tile += NWAVE) {
    const int n = tile * 16 + l15;
    v8f c = {};
#pragma unroll
    for (int i = 0; i < KS; ++i) {
      v2f b = wload<WHICH>(Wa, Wb, n, i * 4 + ks);
      c = __builtin_amdgcn_wmma_f32_16x16x4_f32(false, areg[i], false, b, (short)0, c, false, false);
    }
    float bn;
    if (WHICH == 3) bn = (n < 26) ? ba[n] : ((n < 92) ? bb[n - 26] : 0.f);
    else            bn = (n < 204) ? (ba[n] + bb[n]) : 0.f;
    const int mb = (lane < 16) ? 0 : 8;
#pragma unroll
    for (int r = 0; r < 8; ++r)
      outp[(mb + r) * outld + tile * 16 + l15] = c[r] + bn;
  }
}

// ---------------------------------------------------------------------------
// NTM addressing: half-wave (16 lanes) per batch element, 8 slots per lane.
// ---------------------------------------------------------------------------
__device__ __forceinline__ void address(int m, int l16, const float* o,
                                        float (*w)[NMEM], float (*wg)[NMEM],
                                        float (*mem)[NMEM][MMEM]) {
  float kv[MMEM];
  float k2 = 0.f;
#pragma unroll
  for (int i = 0; i < MMEM; ++i) { kv[i] = o[i]; k2 += kv[i] * kv[i]; }
  const float normK = sqrtf(k2);
  const float beta  = softplusf(o[20]);
  const float g     = sigm(o[21]);
  const float s0r = o[22], s1r = o[23], s2r = o[24];
  const float smx = fmaxf(s0r, fmaxf(s1r, s2r));
  const float e0 = expf(s0r - smx), e1 = expf(s1r - smx), e2 = expf(s2r - smx);
  const float esum = e0 + e1 + e2;
  const float s0 = e0 / esum, s1 = e1 / esum, s2 = e2 / esum;
  const float gamma = 1.f + softplusf(o[25]);

  float sim[8];
  float mx = -1e30f;
#pragma unroll
  for (int j = 0; j < 8; ++j) {
    const int n = l16 + 16 * j;
    const float* mr = mem[m][n];
    float dot = 0.f, nm = 0.f;
#pragma unroll
    for (int mm = 0; mm < MMEM; ++mm) { const float v = mr[mm]; dot += v * kv[mm]; nm += v * v; }
    sim[j] = beta * dot / (sqrtf(nm) * normK + EPSF);
    mx = fmaxf(mx, sim[j]);
  }
  for (int off = 1; off < 16; off <<= 1) mx = fmaxf(mx, __shfl_xor(mx, off, 32));
  float ex[8]; float sum = 0.f;
#pragma unroll
  for (int j = 0; j < 8; ++j) { ex[j] = expf(sim[j] - mx); sum += ex[j]; }
  for (int off = 1; off < 16; off <<= 1) sum += __shfl_xor(sum, off, 32);
  const float inv = 1.f / sum;
#pragma unroll
  for (int j = 0; j < 8; ++j) {
    const int n = l16 + 16 * j;
    wg[m][n] = g * (ex[j] * inv) + (1.f - g) * w[m][n];
  }
  __syncthreads();
  float wp[8]; float ps = 0.f;
#pragma unroll
  for (int j = 0; j < 8; ++j) {
    const int n = l16 + 16 * j;
    const float wt = s0 * wg[m][(n + NMEM - 1) & (NMEM - 1)]
                   + s1 * wg[m][n]
                   + s2 * wg[m][(n + 1) & (NMEM - 1)];
    wp[j] = powf(wt, gamma);
    ps += wp[j];
  }
  for (int off = 1; off < 16; off <<= 1) ps += __shfl_xor(ps, off, 32);
  const float invp = 1.f / (ps + EPSF);
#pragma unroll
  for (int j = 0; j < 8; ++j) w[m][l16 + 16 * j] = wp[j] * invp;
  __syncthreads();
}

// ---------------------------------------------------------------------------
// Persistent per-block NTM scan: one block = 16 batch rows for all T=64 steps.
// ---------------------------------------------------------------------------
template <bool PACKED>
__global__ __launch_bounds__(TPB) void ntm_kernel(
    const float* __restrict__ x,
    const float* __restrict__ Wih1, const float* __restrict__ Whh1,
    const float* __restrict__ bih1, const float* __restrict__ bhh1,
    const float* __restrict__ Wih2, const float* __restrict__ Whh2,
    const float* __restrict__ bih2, const float* __restrict__ bhh2,
    const float* __restrict__ Wread, const float* __restrict__ bread,
    const float* __restrict__ Wwrite, const float* __restrict__ bwrite,
    const float* __restrict__ Wout, const float* __restrict__ bout,
    const float* __restrict__ memInit,
    const float* __restrict__ ws,
    float* __restrict__ out) {
  __shared__ float sMem[BT][NMEM][MMEM];          // 160 KB NTM memory tile
  __shared__ float sAct[BT][120];
  __shared__ float sG[BT][208];                   // gates (aliased as wg scratch)
  __shared__ float sO[BT][96];
  __shared__ float sH1[BT][52], sC1[BT][52], sH2[BT][52], sC2[BT][52];
  __shared__ float sWr[BT][NMEM], sWw[BT][NMEM];
  __shared__ float sR[BT][MMEM];

  float (*sWg)[NMEM] = (float (*)[NMEM]) & sG[0][0];

  const int tid  = threadIdx.x;
  const int wave = tid >> 5;
  const int lane = tid & 31;
  const int b0   = blockIdx.x * BT;
  const int mAddr = wave * 2 + (lane >> 4);
  const int l16   = lane & 15;

  for (int i = tid; i < BT * 52; i += TPB) {
    const int m = i / 52, j = i % 52;
    sH1[m][j] = 0.f; sC1[m][j] = 0.f; sH2[m][j] = 0.f; sC2[m][j] = 0.f;
  }
  for (int i = tid; i < BT * NMEM; i += TPB) {
    const int m = i / NMEM, n = i % NMEM;
    sWr[m][n] = 1.f / NMEM; sWw[m][n] = 1.f / NMEM;
  }
  for (int i = tid; i < NMEM * MMEM; i += TPB) {
    const float v = memInit[i];
    const int n = i / MMEM, mm = i % MMEM;
    for (int m = 0; m < BT; ++m) sMem[m][n][mm] = v;
  }
  for (int i = tid; i < BT * 120; i += TPB) sAct[i / 120][i % 120] = 0.f;
  __syncthreads();

  for (int t = 0; t < TT; ++t) {
    // 1) act1 = [x_t | h1_prev]
    {
      const int m = tid >> 4;
      const int k4 = (tid & 15) * 4;
      const float* xp = &x[((size_t)(b0 + m) * TT + t) * INW + k4];
      sAct[m][k4 + 0] = xp[0]; sAct[m][k4 + 1] = xp[1];
      sAct[m][k4 + 2] = xp[2]; sAct[m][k4 + 3] = xp[3];
    }
    for (int i = tid; i < BT * 52; i += TPB) { const int m = i / 52, j = i % 52; sAct[m][64 + j] = sH1[m][j]; }
    __syncthreads();

    // 2) layer-1 gates
    if constexpr (PACKED)
      gemm_packed<G1_KS>(&sAct[0][0], ws, ws + OFF_B1, G1_TILES, &sG[0][0], 208, wave, lane);
    else
      gemm_guarded<1, 116>(&sAct[0][0], G1_TILES, Wih1, Whh1, bih1, bhh1, &sG[0][0], 208, wave, lane);
    __syncthreads();

    // 3) LSTM cell 1
    for (int i = tid; i < BT * HH; i += TPB) {
      const int m = i / HH, j = i % HH;
      const float ig = sigm(sG[m][j]), fg = sigm(sG[m][51 + j]);
      const float gg = tanhf(sG[m][102 + j]), og = sigm(sG[m][153 + j]);
      const float c = fg * sC1[m][j] + ig * gg;
      sC1[m][j] = c; sH1[m][j] = og * tanhf(c);
    }
    __syncthreads();

    // 4) act2 = [h1 | h2_prev]
    for (int i = tid; i < BT * 104; i += TPB) {
      const int m = i / 104, j = i % 104;
      sAct[m][j] = (j < 52) ? sH1[m][j] : sH2[m][j - 52];
    }
    __syncthreads();

    // 5) layer-2 gates
    if constexpr (PACKED)
      gemm_packed<G2_KS>(&sAct[0][0], ws + OFF_G2, ws + OFF_B2, G2_TILES, &sG[0][0], 208, wave, lane);
    else
      gemm_guarded<2, 104>(&sAct[0][0], G2_TILES, Wih2, Whh2, bih2, bhh2, &sG[0][0], 208, wave, lane);
    __syncthreads();

    // 6) LSTM cell 2
    for (int i = tid; i < BT * HH; i += TPB) {
      const int m = i / HH, j = i % HH;
      const float ig = sigm(sG[m][j]), fg = sigm(sG[m][51 + j]);
      const float gg = tanhf(sG[m][102 + j]), og = sigm(sG[m][153 + j]);
      const float c = fg * sC2[m][j] + ig * gg;
      sC2[m][j] = c; sH2[m][j] = og * tanhf(c);
    }
    __syncthreads();

    // 7) act3 = [h2]
    for (int i = tid; i < BT * 52; i += TPB) { const int m = i / 52, j = i % 52; sAct[m][j] = sH2[m][j]; }
    __syncthreads();

    // 8) head projections -> sO[.][0..91]
    if constexpr (PACKED)
      gemm_packed<GH_KS>(&sAct[0][0], ws + OFF_GH, ws + OFF_BH, GH_TILES, &sO[0][0], 96, wave, lane);
    else
      gemm_guarded<3, 52>(&sAct[0][0], GH_TILES, Wread, Wwrite, bread, bwrite, &sO[0][0], 96, wave, lane);
    __syncthreads();

    // 9) read-head addressing (pre-update mem)
    address(mAddr, l16, &sO[mAddr][0], sWr, sWg, sMem);

    // 10) r = w_r @ mem
    for (int i = tid; i < BT * MMEM; i += TPB) {
      const int m = i / MMEM, mm = i % MMEM;
      float acc = 0.f;
      for (int n = 0; n < NMEM; ++n) acc += sWr[m][n] * sMem[m][n][mm];
      sR[m][mm] = acc;
    }
    __syncthreads();

    // 11) write-head addressing (pre-update mem)
    address(mAddr, l16, &sO[mAddr][26], sWw, sWg, sMem);

    // 12) memory erase/add
    for (int i = tid; i < BT * NMEM * MMEM; i += TPB) {
      const int m = i / (NMEM * MMEM);
      const int r2 = i % (NMEM * MMEM);
      const int n = r2 / MMEM, mm = r2 % MMEM;
      const float ww = sWw[m][n];
      const float e  = sigm(sO[m][52 + mm]);
      const float a  = sO[m][72 + mm];
      const float v  = sMem[m][n][mm];
      sMem[m][n][mm] = v * (1.f - ww * e) + ww * a;
    }

    // 13) output scalar
    if (tid < BT) {
      const int m = tid;
      float acc = bout[0];
      for (int j = 0; j < HH; ++j)      acc += sH2[m][j] * Wout[j];
      for (int mm = 0; mm < MMEM; ++mm) acc += sR[m][mm] * Wout[HH + mm];
      out[(size_t)(b0 + m) * TT + t] = sigm(acc);
    }
    __syncthreads();
  }
}

extern "C" void kernel_launch(void* const* d_in, const int* in_sizes, int n_in,
                              void* d_out, int out_size, void* d_ws, size_t ws_size,
                              hipStream_t stream) {
  (void)in_sizes; (void)n_in; (void)out_size;
  const float* x      = (const float*)d_in[0];
  const float* Wih1   = (const float*)d_in[1];
  const float* Whh1   = (const float*)d_in[2];
  const float* bih1   = (const float*)d_in[3];
  const float* bhh1   = (const float*)d_in[4];
  const float* Wih2   = (const float*)d_in[5];
  const float* Whh2   = (const float*)d_in[6];
  const float* bih2   = (const float*)d_in[7];
  const float* bhh2   = (const float*)d_in[8];
  const float* Wread  = (const float*)d_in[9];
  const float* bread  = (const float*)d_in[10];
  const float* Wwrite = (const float*)d_in[11];
  const float* bwrite = (const float*)d_in[12];
  const float* Wout   = (const float*)d_in[13];
  const float* bout   = (const float*)d_in[14];
  const float* memini = (const float*)d_in[15];
  float* out = (float*)d_out;

  dim3 grid(BATCH / BT), block(TPB);
  const size_t need = (size_t)WS_FLOATS * sizeof(float);

  if (ws_size >= need) {
    float* ws = (float*)d_ws;
    hipLaunchKernelGGL(pack_weights, dim3((WS_FLOATS + 255) / 256), dim3(256), 0, stream,
                       Wih1, Whh1, Wih2, Whh2, Wread, Wwrite,
                       bih1, bhh1, bih2, bhh2, bread, bwrite, ws);
    hipLaunchKernelGGL((ntm_kernel<true>), grid, block, 0, stream,
                       x, Wih1, Whh1, bih1, bhh1, Wih2, Whh2, bih2, bhh2,
                       Wread, bread, Wwrite, bwrite, Wout, bout, memini,
                       (const float*)ws, out);
  } else {
    hipLaunchKernelGGL((ntm_kernel<false>), grid, block, 0, stream,
                       x, Wih1, Whh1, bih1, bhh1, Wih2, Whh2, bih2, bhh2,
                       Wread, bread, Wwrite, bwrite, Wout, bout, memini,
                       (const float*)nullptr, out);
  }
}